// SMPL_22101901705574
// MI455X (gfx1250) — compile-verified
//
#include <hip/hip_runtime.h>
#include <math.h>

// ---------------- problem constants ----------------
#define BATCH 1024
#define NVERT 6890
#define NJ    24
#define NB    10
#define MROWS (NVERT * 3)          // 20670
#define MP    20672                // MROWS padded to multiple of 16
#define KPOSE 207
#define KPOSEP 208                 // padded to multiple of 4
#define KSHAPE 10
#define KSHAPEP 12                 // padded to multiple of 4

typedef __attribute__((ext_vector_type(2))) float v2f;
typedef __attribute__((ext_vector_type(8))) float v8f;

__device__ __constant__ int c_parents[NJ] = {
    -1, 0, 0, 0, 1, 2, 3, 4, 5, 6, 7, 8, 9, 9, 9, 12, 13, 14, 16, 17, 18, 19, 20, 21
};

// ---------------------------------------------------------------------------
// Stage posedirs [M,207] -> zero-padded transpose PDT [208, MP]
// ---------------------------------------------------------------------------
__global__ __launch_bounds__(256)
void stage_pdt_kernel(const float* __restrict__ PD, float* __restrict__ PDT) {
    int idx = blockIdx.x * 256 + threadIdx.x;          // over KPOSEP * MP
    int k = idx / MP;
    int m = idx % MP;
    float v = 0.0f;
    if (k < KPOSE && m < MROWS) v = PD[(size_t)m * KPOSE + k];
    PDT[idx] = v;
}

// Stage shapedirs [M,10] -> SDT [12, MP]; v_template [M] -> VTP [MP]
__global__ __launch_bounds__(256)
void stage_sdt_kernel(const float* __restrict__ SD, const float* __restrict__ VT,
                      float* __restrict__ SDT, float* __restrict__ VTP) {
    int idx = blockIdx.x * 256 + threadIdx.x;          // over KSHAPEP * MP
    int k = idx / MP;
    int m = idx % MP;
    float v = 0.0f;
    if (k < KSHAPE && m < MROWS) v = SD[(size_t)m * KSHAPE + k];
    SDT[idx] = v;
    if (idx < MP) VTP[idx] = (idx < MROWS) ? VT[idx] : 0.0f;
}

// ---------------------------------------------------------------------------
// Rodrigues rotations + transposed pose-feature (pfT [208, B]) + betasT [12,B]
// ---------------------------------------------------------------------------
__global__ __launch_bounds__(256)
void pose_kernel(const float* __restrict__ pose, const float* __restrict__ betas,
                 float* __restrict__ R, float* __restrict__ pfT,
                 float* __restrict__ betasT) {
    int idx = blockIdx.x * 256 + threadIdx.x;          // B*NJ
    if (idx >= BATCH * NJ) return;
    int b = idx / NJ;
    int j = idx % NJ;
    float rx = pose[idx * 3 + 0];
    float ry = pose[idx * 3 + 1];
    float rz = pose[idx * 3 + 2];
    float th = sqrtf(rx * rx + ry * ry + rz * rz) + 1e-8f;
    float inv = 1.0f / th;
    float kx = rx * inv, ky = ry * inv, kz = rz * inv;
    float ct = cosf(th), st = sinf(th), omc = 1.0f - ct;
    float r9[9];
    r9[0] = ct + omc * kx * kx;      r9[1] = omc * kx * ky - st * kz;  r9[2] = omc * kx * kz + st * ky;
    r9[3] = omc * ky * kx + st * kz; r9[4] = ct + omc * ky * ky;       r9[5] = omc * ky * kz - st * kx;
    r9[6] = omc * kz * kx - st * ky; r9[7] = omc * kz * ky + st * kx;  r9[8] = ct + omc * kz * kz;
#pragma unroll
    for (int e = 0; e < 9; ++e) R[idx * 9 + e] = r9[e];
    if (j >= 1) {
        int k0 = (j - 1) * 9;
#pragma unroll
        for (int e = 0; e < 9; ++e) {
            float id = (e == 0 || e == 4 || e == 8) ? 1.0f : 0.0f;
            pfT[(size_t)(k0 + e) * BATCH + b] = r9[e] - id;
        }
    } else {
        pfT[(size_t)KPOSE * BATCH + b] = 0.0f;         // zero pad row 207
#pragma unroll
        for (int l = 0; l < KSHAPEP; ++l)
            betasT[(size_t)l * BATCH + b] = (l < NB) ? betas[b * NB + l] : 0.0f;
    }
}

// ---------------------------------------------------------------------------
// Fold J_regressor through shapedirs/v_template once:
//   JS[j][d][l] = sum_v Jreg[j,v]*shapedirs[v,d,l]; J0[j][d] = sum_v Jreg[j,v]*vt[v,d]
// ---------------------------------------------------------------------------
__global__ __launch_bounds__(256)
void js_kernel(const float* __restrict__ Jreg, const float* __restrict__ SD,
               const float* __restrict__ VT, float* __restrict__ JS,
               float* __restrict__ J0) {
    int j = blockIdx.x;                                // 24 blocks
    int tid = threadIdx.x;
    float accJS[30];
    float acc0[3];
#pragma unroll
    for (int i = 0; i < 30; ++i) accJS[i] = 0.0f;
    acc0[0] = acc0[1] = acc0[2] = 0.0f;
    for (int v = tid; v < NVERT; v += 256) {
        float w = Jreg[(size_t)j * NVERT + v];
#pragma unroll
        for (int d = 0; d < 3; ++d) {
            acc0[d] += w * VT[v * 3 + d];
            const float* sd = &SD[(size_t)(v * 3 + d) * NB];
#pragma unroll
            for (int l = 0; l < NB; ++l) accJS[d * NB + l] += w * sd[l];
        }
    }
    __shared__ float red[256];
    for (int o = 0; o < 33; ++o) {
        red[tid] = (o < 30) ? accJS[o] : acc0[o - 30];
        __syncthreads();
        for (int s = 128; s > 0; s >>= 1) {
            if (tid < s) red[tid] += red[tid + s];
            __syncthreads();
        }
        if (tid == 0) {
            if (o < 30) JS[j * 30 + o] = red[0];
            else        J0[j * 3 + (o - 30)] = red[0];
        }
        __syncthreads();
    }
}

// ---------------------------------------------------------------------------
// Blend GEMM with fp32 WMMA:  VP[b, m] = VTP[m] + SDT^T*betasT + PDT^T*pfT
//   grid = (MP/64, B/64), block = 128 (4 waves).
//   Each wave: one 16-row m-tile x four 16-col b-tiles; the A operand is
//   loaded once per K-step and reused across 4 WMMAs (register blocking).
// ---------------------------------------------------------------------------
__global__ __launch_bounds__(128)
void blend_gemm_kernel(const float* __restrict__ SDT, const float* __restrict__ BT,
                       const float* __restrict__ PDT, const float* __restrict__ PFT,
                       const float* __restrict__ VTP, float* __restrict__ VP) {
    const int wave = threadIdx.x >> 5;
    const int lane = threadIdx.x & 31;
    const int half = lane >> 4;
    const int l15  = lane & 15;
    const int m0 = (blockIdx.x * 4 + wave) * 16;
    const int b0 = blockIdx.y * 64;

    v8f acc[4];
#pragma unroll
    for (int t = 0; t < 4; ++t) acc[t] = (v8f){0.f,0.f,0.f,0.f,0.f,0.f,0.f,0.f};

    // ---- shape blend: K = 12 (padded) ----
#pragma unroll
    for (int kk = 0; kk < KSHAPEP; kk += 4) {
        const int k0 = kk + 2 * half;
        v2f a;
        a.x = SDT[(size_t)(k0    ) * MP + m0 + l15];
        a.y = SDT[(size_t)(k0 + 1) * MP + m0 + l15];
        const float* br0 = &BT[(size_t)(k0    ) * BATCH + b0 + l15];
        const float* br1 = &BT[(size_t)(k0 + 1) * BATCH + b0 + l15];
#pragma unroll
        for (int t = 0; t < 4; ++t) {
            v2f bb;
            bb.x = br0[t * 16];
            bb.y = br1[t * 16];
            acc[t] = __builtin_amdgcn_wmma_f32_16x16x4_f32(false, a, false, bb,
                                                           (short)0, acc[t], false, false);
        }
    }
    // ---- pose blend: K = 208 (padded) ----
    for (int kk = 0; kk < KPOSEP; kk += 4) {
        const int k0 = kk + 2 * half;
        v2f a;
        a.x = PDT[(size_t)(k0    ) * MP + m0 + l15];
        a.y = PDT[(size_t)(k0 + 1) * MP + m0 + l15];
        const float* br0 = &PFT[(size_t)(k0    ) * BATCH + b0 + l15];
        const float* br1 = &PFT[(size_t)(k0 + 1) * BATCH + b0 + l15];
#pragma unroll
        for (int t = 0; t < 4; ++t) {
            v2f bb;
            bb.x = br0[t * 16];
            bb.y = br1[t * 16];
            acc[t] = __builtin_amdgcn_wmma_f32_16x16x4_f32(false, a, false, bb,
                                                           (short)0, acc[t], false, false);
        }
    }

    // add v_template and transpose each 16x16 sub-tile through LDS -> b-major
    __shared__ float tile[4][16 * 17];
    float* tl = tile[wave];
    const int brow = lane >> 1;
    const int mh = (lane & 1) * 8;
#pragma unroll
    for (int t = 0; t < 4; ++t) {
#pragma unroll
        for (int r = 0; r < 8; ++r) {
            int mrow = r + half * 8;
            tl[l15 * 17 + mrow] = acc[t][r] + VTP[m0 + mrow];   // tile[bcol][mrow]
        }
        __syncthreads();
        const float* src = &tl[brow * 17 + mh];
        float* dst = &VP[(size_t)(b0 + t * 16 + brow) * MP + m0 + mh];
#pragma unroll
        for (int i = 0; i < 8; ++i) dst[i] = src[i];
        __syncthreads();
    }
}

// ---------------------------------------------------------------------------
// Kinematic chain: one thread per batch element -> G12 [B, 24, 12] (3x4 rows)
// ---------------------------------------------------------------------------
__global__ __launch_bounds__(256)
void chain_kernel(const float* __restrict__ R, const float* __restrict__ JS,
                  const float* __restrict__ J0, const float* __restrict__ betas,
                  float* __restrict__ G12) {
    int b = blockIdx.x * 256 + threadIdx.x;
    if (b >= BATCH) return;
    float J[NJ][3];
#pragma unroll 4
    for (int i = 0; i < NJ; ++i)
#pragma unroll
        for (int d = 0; d < 3; ++d) {
            float s = J0[i * 3 + d];
            const float* js = &JS[(i * 3 + d) * NB];
#pragma unroll
            for (int l = 0; l < NB; ++l) s += js[l] * betas[b * NB + l];
            J[i][d] = s;
        }
    float Gr[NJ][9];
    float Gt[NJ][3];
    const float* r0 = &R[(size_t)(b * NJ + 0) * 9];
#pragma unroll
    for (int e = 0; e < 9; ++e) Gr[0][e] = r0[e];
#pragma unroll
    for (int d = 0; d < 3; ++d) Gt[0][d] = J[0][d];
    for (int i = 1; i < NJ; ++i) {
        int p = c_parents[i];
        const float* ri = &R[(size_t)(b * NJ + i) * 9];
        float t[3] = {J[i][0] - J[p][0], J[i][1] - J[p][1], J[i][2] - J[p][2]};
#pragma unroll
        for (int row = 0; row < 3; ++row) {
            float g0 = Gr[p][row * 3 + 0], g1 = Gr[p][row * 3 + 1], g2 = Gr[p][row * 3 + 2];
#pragma unroll
            for (int col = 0; col < 3; ++col)
                Gr[i][row * 3 + col] = g0 * ri[col] + g1 * ri[3 + col] + g2 * ri[6 + col];
            Gt[i][row] = g0 * t[0] + g1 * t[1] + g2 * t[2] + Gt[p][row];
        }
    }
#pragma unroll 4
    for (int i = 0; i < NJ; ++i) {
        float* g = &G12[(size_t)(b * NJ + i) * 12];
#pragma unroll
        for (int row = 0; row < 3; ++row) {
            float tcor = Gt[i][row] - (Gr[i][row * 3 + 0] * J[i][0] +
                                       Gr[i][row * 3 + 1] * J[i][1] +
                                       Gr[i][row * 3 + 2] * J[i][2]);
            g[row * 4 + 0] = Gr[i][row * 3 + 0];
            g[row * 4 + 1] = Gr[i][row * 3 + 1];
            g[row * 4 + 2] = Gr[i][row * 3 + 2];
            g[row * 4 + 3] = tcor;
        }
    }
}

// ---------------------------------------------------------------------------
// Fused LBS: verts[b,v,:] = sum_n w[v,n]*(G12[b,n] @ [vp,1]) + trans[b]
//   block = 256 (8 waves, one b per wave), grid = (B/8, vchunks)
// ---------------------------------------------------------------------------
#define VCHUNK 431
__global__ __launch_bounds__(256)
void lbs_kernel(const float* __restrict__ VP, const float* __restrict__ W,
                const float* __restrict__ G12, const float* __restrict__ trans,
                float* __restrict__ verts) {
    __shared__ float sg[8][NJ * 12];
    int tid = threadIdx.x;
    int bbase = blockIdx.x * 8;
    for (int i = tid; i < 8 * NJ * 12; i += 256) {
        int w8 = i / (NJ * 12);
        int e = i % (NJ * 12);
        sg[w8][e] = G12[(size_t)(bbase + w8) * NJ * 12 + e];
    }
    __syncthreads();
    int wave = tid >> 5;
    int lane = tid & 31;
    int b = bbase + wave;
    float tx = trans[b * 3 + 0], ty = trans[b * 3 + 1], tz = trans[b * 3 + 2];
    const float* vp = &VP[(size_t)b * MP];
    const float* gw = sg[wave];
    int v0 = blockIdx.y * VCHUNK;
    int vend = v0 + VCHUNK;
    if (vend > NVERT) vend = NVERT;
    for (int v = v0 + lane; v < vend; v += 32) {
        float px = vp[v * 3 + 0];
        float py = vp[v * 3 + 1];
        float pz = vp[v * 3 + 2];
        float ax = 0.f, ay = 0.f, az = 0.f;
        const float* wrow = &W[(size_t)v * NJ];
#pragma unroll 4
        for (int n = 0; n < NJ; ++n) {
            float w = wrow[n];
            const float* g = &gw[n * 12];
            ax += w * (g[0] * px + g[1] * py + g[2]  * pz + g[3]);
            ay += w * (g[4] * px + g[5] * py + g[6]  * pz + g[7]);
            az += w * (g[8] * px + g[9] * py + g[10] * pz + g[11]);
        }
        float* o = &verts[(size_t)b * NVERT * 3 + (size_t)v * 3];
        o[0] = ax + tx;
        o[1] = ay + ty;
        o[2] = az + tz;
    }
}

// ---------------------------------------------------------------------------
// Final joint regression: one wave per (b, j)
// ---------------------------------------------------------------------------
__global__ __launch_bounds__(256)
void joints_kernel(const float* __restrict__ jr, const float* __restrict__ verts,
                   float* __restrict__ joints) {
    int gw = blockIdx.x * 8 + (threadIdx.x >> 5);      // over B*NJ waves
    int lane = threadIdx.x & 31;
    int b = gw / NJ;
    int j = gw % NJ;
    const float* vb = &verts[(size_t)b * NVERT * 3];
    const float* jrow = &jr[(size_t)j * NVERT];
    float ax = 0.f, ay = 0.f, az = 0.f;
    for (int v = lane; v < NVERT; v += 32) {
        float w = jrow[v];
        ax += w * vb[v * 3 + 0];
        ay += w * vb[v * 3 + 1];
        az += w * vb[v * 3 + 2];
    }
#pragma unroll
    for (int off = 16; off > 0; off >>= 1) {
        ax += __shfl_down(ax, off, 32);
        ay += __shfl_down(ay, off, 32);
        az += __shfl_down(az, off, 32);
    }
    if (lane == 0) {
        joints[(size_t)b * NJ * 3 + j * 3 + 0] = ax;
        joints[(size_t)b * NJ * 3 + j * 3 + 1] = ay;
        joints[(size_t)b * NJ * 3 + j * 3 + 2] = az;
    }
}

// ---------------------------------------------------------------------------
extern "C" void kernel_launch(void* const* d_in, const int* in_sizes, int n_in,
                              void* d_out, int out_size, void* d_ws, size_t ws_size,
                              hipStream_t stream) {
    const float* betas      = (const float*)d_in[0];
    const float* pose       = (const float*)d_in[1];
    const float* trans      = (const float*)d_in[2];
    const float* v_template = (const float*)d_in[3];
    const float* shapedirs  = (const float*)d_in[4];
    const float* posedirs   = (const float*)d_in[5];
    const float* Jreg       = (const float*)d_in[6];
    const float* weights    = (const float*)d_in[7];
    const float* jointreg   = (const float*)d_in[8];

    float* verts  = (float*)d_out;
    float* joints = verts + (size_t)BATCH * NVERT * 3;

    // workspace partition (all fp32)
    float* w = (float*)d_ws;
    float* PDT = w;                     w += (size_t)KPOSEP * MP;      // 208 x MP
    float* SDT = w;                     w += (size_t)KSHAPEP * MP;     // 12 x MP
    float* VTP = w;                     w += MP;
    float* PFT = w;                     w += (size_t)KPOSEP * BATCH;   // 208 x B
    float* BT  = w;                     w += (size_t)KSHAPEP * BATCH;  // 12 x B
    float* R   = w;                     w += (size_t)BATCH * NJ * 9;
    float* JS  = w;                     w += NJ * 30;
    float* J0  = w;                     w += NJ * 3;
    float* G12 = w;                     w += (size_t)BATCH * NJ * 12;
    float* VP  = w;                     w += (size_t)BATCH * MP;       // v_posed, b-major

    stage_pdt_kernel<<<(KPOSEP * MP) / 256, 256, 0, stream>>>(posedirs, PDT);
    stage_sdt_kernel<<<(KSHAPEP * MP) / 256, 256, 0, stream>>>(shapedirs, v_template, SDT, VTP);
    pose_kernel<<<(BATCH * NJ) / 256, 256, 0, stream>>>(pose, betas, R, PFT, BT);
    js_kernel<<<NJ, 256, 0, stream>>>(Jreg, shapedirs, v_template, JS, J0);
    blend_gemm_kernel<<<dim3(MP / 64, BATCH / 64), 128, 0, stream>>>(SDT, BT, PDT, PFT, VTP, VP);
    chain_kernel<<<BATCH / 256, 256, 0, stream>>>(R, JS, J0, betas, G12);
    lbs_kernel<<<dim3(BATCH / 8, (NVERT + VCHUNK - 1) / VCHUNK), 256, 0, stream>>>(VP, weights, G12, trans, verts);
    joints_kernel<<<(BATCH * NJ) / 8, 256, 0, stream>>>(jointreg, verts, joints);
}